// GCNModel_12927851561251
// MI455X (gfx1250) — compile-verified
//
#include <hip/hip_runtime.h>
#include <hip/hip_bf16.h>

// ---------------------------------------------------------------------------
// GCN: BN -> GCNConv -> ReLU -> GCNConv -> ReLU -> Linear(16) -> Linear(2)
// MI455X (gfx1250): wave32, WMMA f32 16x16x4 for the node-feature GEMMs,
// L2-resident edge gather/scatter with explicit global_atomic_add_f32.
// ---------------------------------------------------------------------------

typedef float v2f __attribute__((ext_vector_type(2)));
typedef float v8f __attribute__((ext_vector_type(8)));

#define DINF 128
#define LDA  132   // 128 + 4 pad -> bank stride 4 (mod 64): conflict-free frags
#define STATS_BLOCKS 256

// Guaranteed single-instruction f32 atomic add, device scope (cross-WGP).
__device__ __forceinline__ void atomic_add_f32_dev(float* p, float v) {
    asm volatile("global_atomic_add_f32 %0, %1, off scope:SCOPE_DEV"
                 :
                 : "v"(p), "v"(v)
                 : "memory");
}

// ---------------- BatchNorm column statistics (deterministic 2-stage) ------
__global__ void stats_partial_kernel(const float* __restrict__ x,
                                     float* __restrict__ psum,
                                     float* __restrict__ psq,
                                     int n, int rows_per_blk) {
    int t  = threadIdx.x;                 // column 0..127
    int r0 = blockIdx.x * rows_per_blk;
    int r1 = r0 + rows_per_blk; if (r1 > n) r1 = n;
    float s = 0.f, q = 0.f;
    for (int r = r0; r < r1; ++r) {
        float v = x[(size_t)r * DINF + t];
        s += v; q += v * v;
    }
    psum[blockIdx.x * DINF + t] = s;
    psq [blockIdx.x * DINF + t] = q;
}

__global__ void stats_final_kernel(const float* __restrict__ psum,
                                   const float* __restrict__ psq,
                                   const float* __restrict__ gamma,
                                   const float* __restrict__ beta,
                                   float* __restrict__ scale,
                                   float* __restrict__ shift,
                                   int n, int nblk) {
    int t = threadIdx.x;                  // 128 threads
    float s = 0.f, q = 0.f;
    for (int b = 0; b < nblk; ++b) { s += psum[b * DINF + t]; q += psq[b * DINF + t]; }
    float inv_n = 1.0f / (float)n;
    float mu   = s * inv_n;
    float var  = q * inv_n - mu * mu;
    float rstd = rsqrtf(var + 1e-5f);
    float sc   = rstd * gamma[t];
    scale[t] = sc;
    shift[t] = beta[t] - mu * sc;
}

// ---------------- degree (integer atomics: deterministic) ------------------
__global__ void deg_kernel(const int* __restrict__ dst, unsigned* __restrict__ deg, int E) {
    int e = blockIdx.x * blockDim.x + threadIdx.x;
    if (e < E) atomicAdd(&deg[dst[e]], 1u);
}

__global__ void dinv_kernel(const unsigned* __restrict__ deg, float* __restrict__ dinv, int n) {
    int i = blockIdx.x * blockDim.x + threadIdx.x;
    if (i < n) dinv[i] = rsqrtf((float)deg[i] + 1.0f);   // +1 self-loop
}

// ---------------- WMMA GEMM: out[n,128] = A[n,128] @ W[128,128]^T ----------
// Block: 128 threads = 4 waves; each wave -> 16 rows x 128 cols.
// A tile (64 rows) staged in LDS (BN optionally fused); B frags from global
// (64 KB weight matrix stays L1/L2-hot).
__global__ void gemm_wmma_kernel(const float* __restrict__ A,
                                 const float* __restrict__ W,
                                 const float* __restrict__ scale,
                                 const float* __restrict__ shift,
                                 int fuse_bn,
                                 float* __restrict__ out, int n) {
    __shared__ float Alds[64 * LDA];
    const int tid     = threadIdx.x;          // 0..127
    const int rowbase = blockIdx.x * 64;

    // Stage 64x128 tile as float4, fully coalesced; zero-pad tail rows.
    for (int idx = tid; idx < 64 * 32; idx += 128) {
        int r  = idx >> 5;
        int c4 = idx & 31;
        int gr = rowbase + r;
        float4 v;
        if (gr < n) {
            v = ((const float4*)(A + (size_t)gr * DINF))[c4];
            if (fuse_bn) {
                float4 sc = ((const float4*)scale)[c4];
                float4 sh = ((const float4*)shift)[c4];
                v.x = v.x * sc.x + sh.x;  v.y = v.y * sc.y + sh.y;
                v.z = v.z * sc.z + sh.z;  v.w = v.w * sc.w + sh.w;
            }
        } else {
            v = make_float4(0.f, 0.f, 0.f, 0.f);
        }
        *(float4*)&Alds[r * LDA + c4 * 4] = v;
    }
    __syncthreads();

    const int lane = tid & 31;
    const int wv   = tid >> 5;                // wave 0..3 -> rows wv*16..+15
    const int m    = lane & 15;               // A/B: M (or N) index
    const int koff = (lane < 16) ? 0 : 2;     // f32 16x4 frag: K split per half

    v8f zero = {};
    v8f acc[8];
#pragma unroll
    for (int i = 0; i < 8; ++i) acc[i] = zero;

    const float* arow = &Alds[(wv * 16 + m) * LDA];
    for (int k = 0; k < DINF; k += 4) {
        v2f a = *(const v2f*)&arow[k + koff];
#pragma unroll
        for (int nb = 0; nb < 8; ++nb) {
            // B[k][j] = W[j][k], j = nb*16 + m -> contiguous float2 in W row.
            v2f b = *(const v2f*)&W[(size_t)(nb * 16 + m) * DINF + k + koff];
            acc[nb] = __builtin_amdgcn_wmma_f32_16x16x4_f32(
                false, a, false, b, (short)0, acc[nb], false, false);
        }
    }

    const int hi = lane >> 4;                 // C/D layout: lanes16-31 -> M+8
#pragma unroll
    for (int nb = 0; nb < 8; ++nb) {
#pragma unroll
        for (int g = 0; g < 8; ++g) {
            int row = rowbase + wv * 16 + g + hi * 8;
            if (row < n) out[(size_t)row * DINF + nb * 16 + m] = acc[nb][g];
        }
    }
}

// ---------------- aggregation ---------------------------------------------
// Init agg with the self-loop message (replaces a memset with useful work).
__global__ void selfloop_kernel(const float* __restrict__ h,
                                const float* __restrict__ dinv,
                                float* __restrict__ agg, int n) {
    int idx = blockIdx.x * blockDim.x + threadIdx.x;   // float4 index
    if (idx >= n * 32) return;
    int i = idx >> 5;
    float dv = dinv[i];
    float w  = dv * dv;
    float4 v = ((const float4*)h)[idx];
    v.x *= w; v.y *= w; v.z *= w; v.w *= w;
    ((float4*)agg)[idx] = v;
}

// One wave per edge: one global_load_b128 per lane gathers the whole 512 B
// src row; 4 hardware f32 atomic adds per lane scatter into the dst row.
// Whole h (51 MB) is L2-resident on MI455X (192 MB L2).
__global__ void agg_edge_kernel(const float* __restrict__ h,
                                const int* __restrict__ src,
                                const int* __restrict__ dst,
                                const float* __restrict__ dinv,
                                float* __restrict__ agg, int E) {
    int lane = threadIdx.x & 31;
    int wid  = blockIdx.x * (blockDim.x >> 5) + (threadIdx.x >> 5);
    if (wid >= E) return;
    int s = src[wid];
    int d = dst[wid];
    float w = dinv[s] * dinv[d];
    float4 v = ((const float4*)(h + (size_t)s * DINF))[lane];   // global_load_b128
    float* ad = agg + (size_t)d * DINF + lane * 4;
    atomic_add_f32_dev(ad + 0, v.x * w);
    atomic_add_f32_dev(ad + 1, v.y * w);
    atomic_add_f32_dev(ad + 2, v.z * w);
    atomic_add_f32_dev(ad + 3, v.w * w);
}

__global__ void bias_relu_kernel(float* __restrict__ buf,
                                 const float* __restrict__ bias, int n) {
    int idx = blockIdx.x * blockDim.x + threadIdx.x;   // float4 index
    if (idx >= n * 32) return;
    int c4 = idx & 31;
    float4 b = ((const float4*)bias)[c4];
    float4 v = ((float4*)buf)[idx];
    v.x = fmaxf(v.x + b.x, 0.f);
    v.y = fmaxf(v.y + b.y, 0.f);
    v.z = fmaxf(v.z + b.z, 0.f);
    v.w = fmaxf(v.w + b.w, 0.f);
    ((float4*)buf)[idx] = v;
}

// ---------------- classifier: relu(h@Wc1^T+bc1)@Wc2^T+bc2 ------------------
__global__ void classifier_kernel(const float* __restrict__ h,
                                  const float* __restrict__ Wc1,
                                  const float* __restrict__ bc1,
                                  const float* __restrict__ Wc2,
                                  const float* __restrict__ bc2,
                                  float* __restrict__ out, int n) {
    int node = blockIdx.x * blockDim.x + threadIdx.x;
    if (node >= n) return;
    float z[16];
#pragma unroll
    for (int j = 0; j < 16; ++j) z[j] = bc1[j];
    const float* row = h + (size_t)node * DINF;
    for (int k = 0; k < DINF; ++k) {
        float v = row[k];
#pragma unroll
        for (int j = 0; j < 16; ++j) z[j] = fmaf(v, Wc1[j * DINF + k], z[j]);  // uniform W -> s_load
    }
    float o0 = bc2[0], o1 = bc2[1];
#pragma unroll
    for (int j = 0; j < 16; ++j) {
        float r = z[j] > 0.f ? z[j] : 0.f;
        o0 = fmaf(r, Wc2[j],      o0);
        o1 = fmaf(r, Wc2[16 + j], o1);
    }
    out[(size_t)node * 2]     = o0;
    out[(size_t)node * 2 + 1] = o1;
}

// ---------------------------------------------------------------------------
extern "C" void kernel_launch(void* const* d_in, const int* in_sizes, int n_in,
                              void* d_out, int out_size, void* d_ws, size_t ws_size,
                              hipStream_t stream) {
    const float* x     = (const float*)d_in[0];
    const int*   eidx  = (const int*)  d_in[1];
    const float* gamma = (const float*)d_in[2];
    const float* beta  = (const float*)d_in[3];
    const float* W1    = (const float*)d_in[4];
    const float* b1    = (const float*)d_in[5];
    const float* W2    = (const float*)d_in[6];
    const float* b2    = (const float*)d_in[7];
    const float* Wc1   = (const float*)d_in[8];
    const float* bc1   = (const float*)d_in[9];
    const float* Wc2   = (const float*)d_in[10];
    const float* bc2   = (const float*)d_in[11];
    float* out = (float*)d_out;

    const int n = in_sizes[0] / DINF;      // 100000
    const int E = in_sizes[1] / 2;         // 1600000
    const int* src = eidx;
    const int* dst = eidx + E;

    // workspace layout
    size_t nf = (size_t)n * DINF;
    float*    buf0  = (float*)d_ws;                 // GEMM output h
    float*    buf1  = buf0 + nf;                    // aggregation / activations
    float*    dinv  = buf1 + nf;
    unsigned* deg   = (unsigned*)(dinv + n);
    float*    psum  = (float*)(deg + n);            // STATS_BLOCKS*128
    float*    psq   = psum + STATS_BLOCKS * DINF;
    float*    scale = psq  + STATS_BLOCKS * DINF;
    float*    shift = scale + DINF;

    // ---- BN stats (deterministic) ----
    int rows_per_blk = (n + STATS_BLOCKS - 1) / STATS_BLOCKS;
    stats_partial_kernel<<<STATS_BLOCKS, 128, 0, stream>>>(x, psum, psq, n, rows_per_blk);
    stats_final_kernel<<<1, 128, 0, stream>>>(psum, psq, gamma, beta, scale, shift, n, STATS_BLOCKS);

    // ---- degrees / dinv ----
    hipMemsetAsync(deg, 0, (size_t)n * sizeof(unsigned), stream);
    deg_kernel<<<(E + 255) / 256, 256, 0, stream>>>(dst, deg, E);
    dinv_kernel<<<(n + 255) / 256, 256, 0, stream>>>(deg, dinv, n);

    const int gemm_grid = (n + 63) / 64;
    const int elt_grid  = (n * 32 + 255) / 256;
    const int edge_grid = (E + 7) / 8;        // 8 waves (256 thr) per block

    // ---- layer 1: h = BN(x) @ W1^T ; agg ; +b1 ; relu ----
    gemm_wmma_kernel<<<gemm_grid, 128, 0, stream>>>(x, W1, scale, shift, 1, buf0, n);
    selfloop_kernel<<<elt_grid, 256, 0, stream>>>(buf0, dinv, buf1, n);
    agg_edge_kernel<<<edge_grid, 256, 0, stream>>>(buf0, src, dst, dinv, buf1, E);
    bias_relu_kernel<<<elt_grid, 256, 0, stream>>>(buf1, b1, n);

    // ---- layer 2 ----
    gemm_wmma_kernel<<<gemm_grid, 128, 0, stream>>>(buf1, W2, nullptr, nullptr, 0, buf0, n);
    selfloop_kernel<<<elt_grid, 256, 0, stream>>>(buf0, dinv, buf1, n);
    agg_edge_kernel<<<edge_grid, 256, 0, stream>>>(buf0, src, dst, dinv, buf1, E);
    bias_relu_kernel<<<elt_grid, 256, 0, stream>>>(buf1, b2, n);

    // ---- classifier ----
    classifier_kernel<<<(n + 127) / 128, 128, 0, stream>>>(buf1, Wc1, bc1, Wc2, bc2, out, n);
}